// ContextMatching_10058813407529
// MI455X (gfx1250) — compile-verified
//
#include <hip/hip_runtime.h>
#include <math.h>

typedef __attribute__((ext_vector_type(2))) float v2f;
typedef __attribute__((ext_vector_type(8))) float v8f;

#define NN 1024
#define CC 128
#define TT 12
#define PP 24
#define HH 24
#define GELEMS (NN * NN)
#define APAD 18

// ---------------------------------------------------------------------------
// K1: sc[t][n][c] = inputs[b][n][t][d],  c = b*2 + d   (12,1024,128) fp32
// ---------------------------------------------------------------------------
__global__ __launch_bounds__(256) void build_sc(const float* __restrict__ in,
                                                float* __restrict__ sc) {
  int i = blockIdx.x * 256 + threadIdx.x;        // exactly 12*1024*128 threads
  int c = i & (CC - 1);
  int rest = i >> 7;
  int n = rest & (NN - 1);
  int t = rest >> 10;
  int b = c >> 1, d = c & 1;
  sc[i] = in[(((size_t)b * NN + n) * TT + t) * 2 + d];
}

// ---------------------------------------------------------------------------
// K0: zero metric sums, gi = broadcast(b_ih)
// ---------------------------------------------------------------------------
__global__ void init_kernel(const float* __restrict__ b_ih,
                            float* __restrict__ msum, float* __restrict__ gi) {
  int i = blockIdx.x * 256 + threadIdx.x;
  if (i < (TT - 1) * PP) msum[i] = 0.f;
  if (i < TT * 72) gi[i] = b_ih[i % 72];
}

// ---------------------------------------------------------------------------
// K2: fused einsum + masked |pred-target| reduction, V_WMMA_F32_16X16X4_F32.
// Each wave computes a 16x64 tile (4 accumulators): one A fragment feeds 4
// WMMAs, so 16 WMMAs per LDS stage per wave (4x fewer barriers per FLOP).
// grid: (c-tiles=2, row-tiles=8, t*24+p=264), 256 thr = 8 waves
// ---------------------------------------------------------------------------
__global__ __launch_bounds__(256) void metrics_kernel(
    const float* __restrict__ graphs, const float* __restrict__ sc,
    float* __restrict__ msum) {
  __shared__ float Alds[128 * APAD];
  __shared__ float Blds[64 * APAD];
  const int tp = blockIdx.z;
  const int t = tp / PP, p = tp % PP;
  const float* G = graphs + (size_t)p * GELEMS;
  const float* X = sc + (size_t)t * NN * CC;         // x1[t]
  const float* Tg = sc + (size_t)(t + 1) * NN * CC;  // target[t]
  const int r0 = blockIdx.y * 128;
  const int c0 = blockIdx.x * 64;
  const int tid = threadIdx.x;
  const int lane = tid & 31;
  const int wave = tid >> 5;
  const int m = lane & 15;
  const int h = lane >> 4;
  const int wr0 = wave * 16;

  v8f acc[4] = {};
  for (int k0 = 0; k0 < NN; k0 += 16) {
    {  // A: 128 rows x 16 k, coalesced float4 pairs
      int row = tid >> 1;
      int off = (tid & 1) * 8;
      const float* src = G + (size_t)(r0 + row) * NN + k0 + off;
      float4 u0 = *reinterpret_cast<const float4*>(src);
      float4 u1 = *reinterpret_cast<const float4*>(src + 4);
      float* dst = &Alds[row * APAD + off];
      dst[0] = u0.x; dst[1] = u0.y; dst[2] = u0.z; dst[3] = u0.w;
      dst[4] = u1.x; dst[5] = u1.y; dst[6] = u1.z; dst[7] = u1.w;
    }
    {  // B stored transposed: Blds[n][k] = X[k0+kr][c0+n], float4-vectorized
      int kr = tid >> 4;
      int ccb = (tid & 15) * 4;
      float4 u = *reinterpret_cast<const float4*>(
          X + (size_t)(k0 + kr) * CC + c0 + ccb);
      Blds[(ccb + 0) * APAD + kr] = u.x;
      Blds[(ccb + 1) * APAD + kr] = u.y;
      Blds[(ccb + 2) * APAD + kr] = u.z;
      Blds[(ccb + 3) * APAD + kr] = u.w;
    }
    __syncthreads();
#pragma unroll
    for (int kk = 0; kk < 16; kk += 4) {
      // A frag (16x4): lane h*16+m -> {A[m][kk+2h], A[m][kk+2h+1]}
      v2f a = *reinterpret_cast<const v2f*>(&Alds[(wr0 + m) * APAD + kk + 2 * h]);
#pragma unroll
      for (int nb = 0; nb < 4; ++nb) {
        // B frag (4x16): lane h*16+n -> {B[kk+2h][n], B[kk+2h+1][n]}
        v2f b = *reinterpret_cast<const v2f*>(
            &Blds[(nb * 16 + m) * APAD + kk + 2 * h]);
        acc[nb] = __builtin_amdgcn_wmma_f32_16x16x4_f32(
            false, a, false, b, (short)0, acc[nb], false, false);
      }
    }
    __syncthreads();
  }
  // C layout: VGPR v, lane h*16+n -> element (M = v + 8h, N = n)
  float s = 0.f;
#pragma unroll
  for (int nb = 0; nb < 4; ++nb) {
#pragma unroll
    for (int v = 0; v < 8; ++v) {
      int row = r0 + wr0 + v + 8 * h;
      int col = c0 + nb * 16 + m;
      float tv = Tg[(size_t)row * CC + col];
      if (tv != 0.f) {
        float l = fabsf(acc[nb][v] - tv);
        if (l == l) s += l;  // NaN guard per reference
      }
    }
  }
  for (int off = 16; off > 0; off >>= 1) s += __shfl_xor(s, off, 32);
  if (lane == 0) atomicAdd(&msum[t * PP + p], s);
}

// ---------------------------------------------------------------------------
// K3: idx[t] = argmin_p msum[t][p]  (mask normalization is a positive per-t
// constant, so argmin of the raw masked sum is identical)
// ---------------------------------------------------------------------------
__global__ void argmin_kernel(const float* __restrict__ msum,
                              int* __restrict__ idx) {
  int t = threadIdx.x;
  if (t < TT - 1) {
    int best = 0;
    float bv = msum[t * PP];
    for (int p = 1; p < PP; ++p) {
      float v = msum[t * PP + p];
      if (v < bv) { bv = v; best = p; }
    }
    idx[t] = best;
  }
}

// ---------------------------------------------------------------------------
// K4a: xm = last - rowmean(last)
// ---------------------------------------------------------------------------
__global__ __launch_bounds__(128) void xm_kernel(const float* __restrict__ sc,
                                                 float* __restrict__ xm) {
  __shared__ float red[128];
  int row = blockIdx.x, tid = threadIdx.x;
  float v = sc[(size_t)(TT - 1) * NN * CC + (size_t)row * CC + tid];
  red[tid] = v;
  __syncthreads();
  for (int o = 64; o > 0; o >>= 1) {
    if (tid < o) red[tid] += red[tid + o];
    __syncthreads();
  }
  xm[(size_t)row * CC + tid] = v - red[0] * (1.f / 128.f);
}

// ---------------------------------------------------------------------------
// K4b: cov = xm @ xm^T / 127 via WMMA f32 16x16x4, K=128, 16x64 wave tiles
// grid: (n-tiles=16, row-tiles=8)
// ---------------------------------------------------------------------------
__global__ __launch_bounds__(256) void cov_kernel(const float* __restrict__ xm,
                                                  float* __restrict__ cov) {
  __shared__ float Alds[128 * APAD];
  __shared__ float Blds[64 * APAD];
  const int r0 = blockIdx.y * 128;
  const int n0 = blockIdx.x * 64;
  const int tid = threadIdx.x;
  const int lane = tid & 31;
  const int wave = tid >> 5;
  const int m = lane & 15;
  const int h = lane >> 4;
  const int wr0 = wave * 16;

  v8f acc[4] = {};
  for (int k0 = 0; k0 < CC; k0 += 16) {
    {
      int row = tid >> 1;
      int off = (tid & 1) * 8;
      const float* src = xm + (size_t)(r0 + row) * CC + k0 + off;
      float4 u0 = *reinterpret_cast<const float4*>(src);
      float4 u1 = *reinterpret_cast<const float4*>(src + 4);
      float* dst = &Alds[row * APAD + off];
      dst[0] = u0.x; dst[1] = u0.y; dst[2] = u0.z; dst[3] = u0.w;
      dst[4] = u1.x; dst[5] = u1.y; dst[6] = u1.z; dst[7] = u1.w;
    }
    {  // B = xm^T: Blds[n][k] = xm[n0+n][k0+k], float4 along k
      int n = tid >> 2;
      int kb = (tid & 3) * 4;
      float4 u = *reinterpret_cast<const float4*>(
          xm + (size_t)(n0 + n) * CC + k0 + kb);
      float* dst = &Blds[n * APAD + kb];
      dst[0] = u.x; dst[1] = u.y; dst[2] = u.z; dst[3] = u.w;
    }
    __syncthreads();
#pragma unroll
    for (int kk = 0; kk < 16; kk += 4) {
      v2f a = *reinterpret_cast<const v2f*>(&Alds[(wr0 + m) * APAD + kk + 2 * h]);
#pragma unroll
      for (int nb = 0; nb < 4; ++nb) {
        v2f b = *reinterpret_cast<const v2f*>(
            &Blds[(nb * 16 + m) * APAD + kk + 2 * h]);
        acc[nb] = __builtin_amdgcn_wmma_f32_16x16x4_f32(
            false, a, false, b, (short)0, acc[nb], false, false);
      }
    }
    __syncthreads();
  }
#pragma unroll
  for (int nb = 0; nb < 4; ++nb) {
#pragma unroll
    for (int v = 0; v < 8; ++v) {
      int row = r0 + wr0 + v + 8 * h;
      cov[(size_t)row * NN + n0 + nb * 16 + m] = acc[nb][v] * (1.f / 127.f);
    }
  }
}

// ---------------------------------------------------------------------------
// K5: gi[t][j] += sum_k ctx[t][k] * W_ih[j][k]  (W_ih = 302 MB, read once;
// one W element feeds all 12 context rows which stay L2-resident)
// grid: (j=72, k-segments=32 of 32768)
// ---------------------------------------------------------------------------
__global__ __launch_bounds__(256) void gi_kernel(
    const float* __restrict__ W_ih, const float* __restrict__ graphs,
    const float* __restrict__ cov, const int* __restrict__ idx,
    float* __restrict__ gi) {
  const int j = blockIdx.x;
  const size_t base = (size_t)blockIdx.y * 32768;
  const int tid = threadIdx.x;
  const float* ctx[TT];
#pragma unroll
  for (int t = 0; t < TT - 1; ++t) ctx[t] = graphs + (size_t)idx[t] * GELEMS;
  ctx[TT - 1] = cov;
  float acc[TT];
#pragma unroll
  for (int t = 0; t < TT; ++t) acc[t] = 0.f;
  const float* Wrow = W_ih + (size_t)j * GELEMS;
  for (size_t k = base + tid; k < base + 32768; k += 256) {
    float w = Wrow[k];
#pragma unroll
    for (int t = 0; t < TT; ++t) acc[t] += w * ctx[t][k];
  }
  __shared__ float red[256];
  for (int t = 0; t < TT; ++t) {
    red[tid] = acc[t];
    __syncthreads();
    for (int o = 128; o > 0; o >>= 1) {
      if (tid < o) red[tid] += red[tid + o];
      __syncthreads();
    }
    if (tid == 0) atomicAdd(&gi[t * 72 + j], red[0]);
    __syncthreads();
  }
}

// ---------------------------------------------------------------------------
// K6: tiny GRU scan (H=24, 12 steps) + argmax (softmax is monotonic)
// ---------------------------------------------------------------------------
__global__ __launch_bounds__(128) void gru_kernel(const float* __restrict__ gi,
                                                  const float* __restrict__ W_hh,
                                                  const float* __restrict__ b_hh,
                                                  int* __restrict__ idx) {
  __shared__ float h[HH], gh[72], hn[HH];
  int tid = threadIdx.x;
  if (tid < HH) h[tid] = 0.f;
  __syncthreads();
  for (int s = 0; s < TT; ++s) {
    const float* g = gi + s * 72;
    if (tid < 72) {
      float a = b_hh[tid];
      for (int k = 0; k < HH; ++k) a += h[k] * W_hh[tid * HH + k];
      gh[tid] = a;
    }
    __syncthreads();
    if (tid < HH) {
      float r = 1.f / (1.f + expf(-(g[tid] + gh[tid])));
      float z = 1.f / (1.f + expf(-(g[HH + tid] + gh[HH + tid])));
      float ng = tanhf(g[2 * HH + tid] + r * gh[2 * HH + tid]);
      hn[tid] = (1.f - z) * ng + z * h[tid];
    }
    __syncthreads();
    if (tid < HH) h[tid] = hn[tid];
    __syncthreads();
  }
  if (tid == 0) {
    int best = 0;
    float bv = h[0];
    for (int i = 1; i < HH; ++i)
      if (h[i] > bv) { bv = h[i]; best = i; }
    idx[TT - 1] = best;
  }
}

// ---------------------------------------------------------------------------
// K7: out[t] = graphs[idx[t]]  (48 MB gather-copy)
// ---------------------------------------------------------------------------
__global__ __launch_bounds__(256) void copy_out(const float* __restrict__ graphs,
                                                const int* __restrict__ idx,
                                                float* __restrict__ out) {
  size_t i = (size_t)blockIdx.x * 256 + threadIdx.x;  // exactly 12*2^20 threads
  int t = (int)(i >> 20);
  size_t e = i & (GELEMS - 1);
  out[i] = graphs[(size_t)idx[t] * GELEMS + e];
}

extern "C" void kernel_launch(void* const* d_in, const int* in_sizes, int n_in,
                              void* d_out, int out_size, void* d_ws,
                              size_t ws_size, hipStream_t stream) {
  (void)in_sizes; (void)n_in; (void)out_size; (void)ws_size;
  const float* inputs = (const float*)d_in[0];
  const float* graphs = (const float*)d_in[1];
  const float* W_ih   = (const float*)d_in[2];
  const float* W_hh   = (const float*)d_in[3];
  const float* b_ih   = (const float*)d_in[4];
  const float* b_hh   = (const float*)d_in[5];
  float* out = (float*)d_out;

  // workspace layout
  float* sc   = (float*)d_ws;                 // 12*1024*128 = 1,572,864 f
  float* xm   = sc + (size_t)TT * NN * CC;    // 131,072 f
  float* cov  = xm + (size_t)NN * CC;         // 1,048,576 f
  float* msum = cov + (size_t)GELEMS;         // 264 f
  float* gi   = msum + (TT - 1) * PP;         // 864 f
  int*   idx  = (int*)(gi + TT * 72);         // 12 i

  build_sc<<<6144, 256, 0, stream>>>(inputs, sc);
  init_kernel<<<4, 256, 0, stream>>>(b_ih, msum, gi);
  metrics_kernel<<<dim3(2, 8, (TT - 1) * PP), 256, 0, stream>>>(graphs, sc, msum);
  argmin_kernel<<<1, 32, 0, stream>>>(msum, idx);
  xm_kernel<<<NN, 128, 0, stream>>>(sc, xm);
  cov_kernel<<<dim3(16, 8, 1), 256, 0, stream>>>(xm, cov);
  gi_kernel<<<dim3(72, 32), 256, 0, stream>>>(W_ih, graphs, cov, idx, gi);
  gru_kernel<<<1, 128, 0, stream>>>(gi, W_hh, b_hh, idx);
  copy_out<<<49152, 256, 0, stream>>>(graphs, idx, out);
}